// GCN_MMAE_30726196035755
// MI455X (gfx1250) — compile-verified
//
#include <hip/hip_runtime.h>
#include <hip/hip_bf16.h>

typedef __attribute__((ext_vector_type(2))) float v2f;
typedef __attribute__((ext_vector_type(8))) float v8f;

#define N_NODES 10000
#define E_EDGES 640000
#define DECD    256
#define NCLS    10

// ------------------------------------------------------------------
// WMMA fp32 GEMM:  C[M,N] = affine(A)[M,K] @ B[K,N] + bias, opt ReLU
//   affine(A)[m][k] = (A[m][k]*a_cscale[k] + a_cshift[k]) * a_rscale[m]
// Block: 256 threads = 8 wave32; block tile 64x128; each wave owns a
// 2x2 grid of 16x16 WMMA tiles (4 v8f accumulators, A/B frags reused 2x).
// K staged in LDS in chunks of 32 -> 32 v_wmma per barrier pair.
// Fast path: fully-interior stages use unguarded float4 (b128) staging.
// ------------------------------------------------------------------
#define BM 64
#define BN 128
#define KT 32
#define LDA_S (KT + 4)    // 36 floats: 16B-aligned rows, conflict-free
#define LDB_S (BN + 4)    // 132 floats

template<bool HAS_BN, bool HAS_RS, bool RELU>
__global__ __launch_bounds__(256)
void gemm_wmma_f32(const float* __restrict__ A, int lda,
                   const float* __restrict__ B, int ldb,
                   const float* __restrict__ bias,
                   float* __restrict__ C, int ldc,
                   int M, int N, int K,
                   const float* __restrict__ a_cscale,
                   const float* __restrict__ a_cshift,
                   const float* __restrict__ a_rscale)
{
    __shared__ __align__(16) float As[BM][LDA_S];
    __shared__ __align__(16) float Bs[KT][LDB_S];

    const int tid  = threadIdx.x;
    const int lane = tid & 31;
    const int wave = tid >> 5;   // 0..7
    const int wm   = wave >> 2;  // 0..1 : rows wm*32 .. wm*32+31
    const int wn   = wave & 3;   // 0..3 : cols wn*32 .. wn*32+31
    const int hi   = lane >> 4;  // 0/1
    const int l16  = lane & 15;

    const int row0 = blockIdx.y * BM;
    const int col0 = blockIdx.x * BN;
    const bool mn_full = (row0 + BM <= M) && (col0 + BN <= N);

    v8f acc00 = {}, acc01 = {}, acc10 = {}, acc11 = {};

    const int amr0 = wm * 32 + l16;       // A rows this lane feeds
    const int amr1 = amr0 + 16;
    const int bnc0 = wn * 32 + l16;       // B cols this lane feeds
    const int bnc1 = bnc0 + 16;

    for (int kb = 0; kb < K; kb += KT) {
        if (mn_full && (kb + KT <= K)) {
            // ---------- fast path: vectorized, unguarded staging ----------
            {   // A tile 64x32 = 512 float4, 2 per thread
                #pragma unroll
                for (int j = 0; j < 2; ++j) {
                    int idx = tid + j * 256;
                    int r = idx >> 3, c4 = (idx & 7) * 4;
                    int gk = kb + c4;
                    float4 v = *(const float4*)&A[(long)(row0 + r) * lda + gk];
                    if (HAS_BN) {
                        float4 s = *(const float4*)&a_cscale[gk];
                        float4 t = *(const float4*)&a_cshift[gk];
                        v.x = v.x * s.x + t.x; v.y = v.y * s.y + t.y;
                        v.z = v.z * s.z + t.z; v.w = v.w * s.w + t.w;
                    }
                    if (HAS_RS) {
                        float rs = a_rscale[row0 + r];
                        v.x *= rs; v.y *= rs; v.z *= rs; v.w *= rs;
                    }
                    *(float4*)&As[r][c4] = v;
                }
            }
            {   // B tile 32x128 = 1024 float4, 4 per thread
                #pragma unroll
                for (int j = 0; j < 4; ++j) {
                    int idx = tid + j * 256;
                    int r = idx >> 5, c4 = (idx & 31) * 4;
                    *(float4*)&Bs[r][c4] =
                        *(const float4*)&B[(long)(kb + r) * ldb + col0 + c4];
                }
            }
        } else {
            // ---------- guarded scalar path (edge blocks / K tail) ----------
            for (int idx = tid; idx < BM * KT; idx += 256) {
                int r = idx >> 5, c = idx & 31;
                int gm = row0 + r, gk = kb + c;
                float v = 0.0f;
                if (gm < M && gk < K) {
                    v = A[(long)gm * lda + gk];
                    if (HAS_BN) v = v * a_cscale[gk] + a_cshift[gk];
                    if (HAS_RS) v *= a_rscale[gm];
                }
                As[r][c] = v;
            }
            for (int idx = tid; idx < KT * BN; idx += 256) {
                int r = idx >> 7, c = idx & 127;
                int gk = kb + r, gn = col0 + c;
                Bs[r][c] = (gk < K && gn < N) ? B[(long)gk * ldb + gn] : 0.0f;
            }
        }
        __syncthreads();

        #pragma unroll
        for (int kk = 0; kk < KT; kk += 4) {
            int k2 = kk + 2 * hi;       // lanes 16..31 take K+2,K+3
            v2f a0, a1, b0, b1;
            a0.x = As[amr0][k2 + 0]; a0.y = As[amr0][k2 + 1];
            a1.x = As[amr1][k2 + 0]; a1.y = As[amr1][k2 + 1];
            b0.x = Bs[k2 + 0][bnc0]; b0.y = Bs[k2 + 1][bnc0];
            b1.x = Bs[k2 + 0][bnc1]; b1.y = Bs[k2 + 1][bnc1];
            acc00 = __builtin_amdgcn_wmma_f32_16x16x4_f32(
                false, a0, false, b0, (short)0, acc00, false, false);
            acc01 = __builtin_amdgcn_wmma_f32_16x16x4_f32(
                false, a0, false, b1, (short)0, acc01, false, false);
            acc10 = __builtin_amdgcn_wmma_f32_16x16x4_f32(
                false, a1, false, b0, (short)0, acc10, false, false);
            acc11 = __builtin_amdgcn_wmma_f32_16x16x4_f32(
                false, a1, false, b1, (short)0, acc11, false, false);
        }
        __syncthreads();
    }

    // --- epilogue: bias (+ReLU); vgpr r -> local M = 8*hi + r ---
    const int gn0 = col0 + bnc0;
    const int gn1 = col0 + bnc1;
    const float bv0 = (bias && gn0 < N) ? bias[gn0] : 0.0f;
    const float bv1 = (bias && gn1 < N) ? bias[gn1] : 0.0f;
    #pragma unroll
    for (int r = 0; r < 8; ++r) {
        int gm0 = row0 + wm * 32 + 8 * hi + r;
        int gm1 = gm0 + 16;
        if (gm0 < M) {
            if (gn0 < N) {
                float v = acc00[r] + bv0;
                if (RELU) v = fmaxf(v, 0.0f);
                C[(long)gm0 * ldc + gn0] = v;
            }
            if (gn1 < N) {
                float v = acc01[r] + bv1;
                if (RELU) v = fmaxf(v, 0.0f);
                C[(long)gm0 * ldc + gn1] = v;
            }
        }
        if (gm1 < M) {
            if (gn0 < N) {
                float v = acc10[r] + bv0;
                if (RELU) v = fmaxf(v, 0.0f);
                C[(long)gm1 * ldc + gn0] = v;
            }
            if (gn1 < N) {
                float v = acc11[r] + bv1;
                if (RELU) v = fmaxf(v, 0.0f);
                C[(long)gm1 * ldc + gn1] = v;
            }
        }
    }
}

// ------------------------------------------------------------------
// Per-column batch stats -> folded BN scale/shift for the next GEMM:
//   scale[c] = g[c] * rsqrt(var + 1e-5), shift[c] = b[c] - mu*scale[c]
// ------------------------------------------------------------------
__global__ __launch_bounds__(256)
void colstats(const float* __restrict__ Z, int rows, int cols,
              const float* __restrict__ g, const float* __restrict__ b,
              float* __restrict__ scale, float* __restrict__ shift)
{
    __shared__ float s_sum[256], s_sq[256];
    const int c = blockIdx.x;
    float sum = 0.f, sq = 0.f;
    for (int i = threadIdx.x; i < rows; i += 256) {
        float v = Z[(long)i * cols + c];
        sum += v; sq += v * v;
    }
    s_sum[threadIdx.x] = sum; s_sq[threadIdx.x] = sq;
    __syncthreads();
    for (int s = 128; s > 0; s >>= 1) {
        if (threadIdx.x < s) {
            s_sum[threadIdx.x] += s_sum[threadIdx.x + s];
            s_sq[threadIdx.x]  += s_sq[threadIdx.x + s];
        }
        __syncthreads();
    }
    if (threadIdx.x == 0) {
        float inv = 1.0f / (float)rows;
        float mu  = s_sum[0] * inv;
        float var = s_sq[0] * inv - mu * mu;       // biased var (ddof=0)
        float s   = g[c] * rsqrtf(var + 1e-5f);
        scale[c] = s;
        shift[c] = b[c] - mu * s;
    }
}

// ------------------------------------------------------------------
// Exact lower-median per column via 4-pass radix select (MSB first).
// ------------------------------------------------------------------
__device__ __forceinline__ unsigned fkey(float f) {
    unsigned u = __float_as_uint(f);
    return u ^ ((u >> 31) ? 0xFFFFFFFFu : 0x80000000u);
}
__device__ __forceinline__ float funkey(unsigned u) {
    u ^= ((u >> 31) ? 0x80000000u : 0xFFFFFFFFu);
    return __uint_as_float(u);
}

__global__ __launch_bounds__(256)
void colmedian(const float* __restrict__ D, int rows, int cols,
               float* __restrict__ med)
{
    __shared__ unsigned hist[256];
    __shared__ unsigned s_prefix, s_want;
    const int c = blockIdx.x;
    if (threadIdx.x == 0) { s_prefix = 0u; s_want = (unsigned)((rows - 1) / 2); }
    unsigned prefmask = 0u;
    __syncthreads();
    for (int shift = 24; shift >= 0; shift -= 8) {
        hist[threadIdx.x] = 0u;
        __syncthreads();
        const unsigned pref = s_prefix;
        for (int i = threadIdx.x; i < rows; i += 256) {
            unsigned u = fkey(D[(long)i * cols + c]);
            if ((u & prefmask) == pref)
                atomicAdd(&hist[(u >> shift) & 255u], 1u);
        }
        __syncthreads();
        if (threadIdx.x == 0) {
            unsigned cum = 0, want = s_want;
            for (int bkt = 0; bkt < 256; ++bkt) {
                unsigned h = hist[bkt];
                if (cum + h > want) {
                    s_prefix = pref | ((unsigned)bkt << shift);
                    s_want   = want - cum;
                    break;
                }
                cum += h;
            }
        }
        prefmask |= (0xFFu << shift);
        __syncthreads();
    }
    if (threadIdx.x == 0) med[c] = funkey(s_prefix);
}

// ------------------------------------------------------------------
// Fused impute (dec row or column median) + 3-modality mean
// ------------------------------------------------------------------
__global__ __launch_bounds__(256)
void impute_sum(const float* __restrict__ d0, const float* __restrict__ d1,
                const float* __restrict__ d2, const float* __restrict__ med,
                const int* __restrict__ r0, const int* __restrict__ r1,
                const int* __restrict__ r2,
                int ns0, int ns1, int ns2, float* __restrict__ nf)
{
    int t = blockIdx.x * 256 + threadIdx.x;
    if (t >= N_NODES * DECD) return;
    int i = t / DECD, c = t % DECD;
    int a0 = r0[i], a1 = r1[i], a2 = r2[i];
    float v0 = (a0 < ns0) ? d0[(long)a0 * DECD + c] : med[c];
    float v1 = (a1 < ns1) ? d1[(long)a1 * DECD + c] : med[256 + c];
    float v2 = (a2 < ns2) ? d2[(long)a2 * DECD + c] : med[512 + c];
    nf[t] = (v0 + v1 + v2) * (1.0f / 3.0f);
}

// ------------------------------------------------------------------
// Graph helpers
// ------------------------------------------------------------------
__global__ void zero_f32(float* __restrict__ p, int n) {
    int t = blockIdx.x * 256 + threadIdx.x;
    if (t < n) p[t] = 0.0f;
}
__global__ void degree_count(const int* __restrict__ idx, int e,
                             float* __restrict__ deg) {
    int t = blockIdx.x * 256 + threadIdx.x;
    if (t < e) atomicAdd(&deg[idx[t]], 1.0f);
}
__global__ void inv_sqrt_clip(const float* __restrict__ deg,
                              float* __restrict__ inv, int n) {
    int t = blockIdx.x * 256 + threadIdx.x;
    if (t < n) inv[t] = rsqrtf(fmaxf(deg[t], 1.0f));
}

// scatter: msg[dst] += x[src] * inv_out[src]   (folds D_out^{-1/2})
// 4 edges/block, 64 float4 lanes per edge; nf & msg are L2-resident.
__global__ __launch_bounds__(256)
void scatter_edges(const float* __restrict__ x, const float* __restrict__ rs,
                   const int* __restrict__ src, const int* __restrict__ dst,
                   float* __restrict__ msg)
{
    int lane = threadIdx.x & 63;
    long e = (long)blockIdx.x * 4 + (threadIdx.x >> 6);
    if (e >= E_EDGES) return;
    int s = src[e], d = dst[e];
    float sc = rs[s];
    float4 v = ((const float4*)(x + (long)s * DECD))[lane];
    float* mp = msg + (long)d * DECD + lane * 4;
    atomicAdd(mp + 0, v.x * sc);
    atomicAdd(mp + 1, v.y * sc);
    atomicAdd(mp + 2, v.z * sc);
    atomicAdd(mp + 3, v.w * sc);
}

// ------------------------------------------------------------------
extern "C" void kernel_launch(void* const* d_in, const int* in_sizes, int n_in,
                              void* d_out, int out_size, void* d_ws, size_t ws_size,
                              hipStream_t stream)
{
    (void)in_sizes; (void)n_in; (void)out_size; (void)ws_size;
    const int NS[3] = {9000, 8500, 8000};
    const int DI[3] = {2000, 1500, 1000};

    // ---- workspace carve-out (all region sizes are multiples of 4 floats)
    float* ws = (float*)d_ws;
    size_t off = 0;
    auto alloc = [&](size_t n) { float* p = ws + off; off += n; return p; };
    float* dec[3];
    dec[0] = alloc((size_t)NS[0] * DECD);
    dec[1] = alloc((size_t)NS[1] * DECD);
    dec[2] = alloc((size_t)NS[2] * DECD);
    float* med  = alloc(3 * 256);
    float* z1   = alloc((size_t)NS[0] * 500);
    float* z2   = alloc((size_t)NS[0] * 128);
    float* sc1  = alloc(500);  float* sh1 = alloc(500);
    float* sc2  = alloc(128);  float* sh2 = alloc(128);
    float* nf   = alloc((size_t)N_NODES * DECD);
    float* msg  = alloc((size_t)N_NODES * DECD);
    float* hbuf = alloc((size_t)N_NODES * DECD);
    float* dego = alloc(N_NODES); float* degi = alloc(N_NODES);
    float* invo = alloc(N_NODES); float* invi = alloc(N_NODES);

    auto cdiv = [](int a, int b) { return (a + b - 1) / b; };

    // ---- 3 autoencoder decodes (GEMM -> stats -> GEMM -> stats -> GEMM -> median)
    for (int m = 0; m < 3; ++m) {
        const float* h    = (const float*)d_in[12 * m + 0];
        const float* e1w  = (const float*)d_in[12 * m + 1];
        const float* e1b  = (const float*)d_in[12 * m + 2];
        const float* bn1g = (const float*)d_in[12 * m + 3];
        const float* bn1b = (const float*)d_in[12 * m + 4];
        const float* e2w  = (const float*)d_in[12 * m + 5];
        const float* e2b  = (const float*)d_in[12 * m + 6];
        const float* bn2g = (const float*)d_in[12 * m + 7];
        const float* bn2b = (const float*)d_in[12 * m + 8];
        const float* dw   = (const float*)d_in[12 * m + 9];
        const float* db   = (const float*)d_in[12 * m + 10];
        const int ns = NS[m], di = DI[m];

        dim3 g1(cdiv(500, BN), cdiv(ns, BM));
        gemm_wmma_f32<false, false, false><<<g1, 256, 0, stream>>>(
            h, di, e1w, 500, e1b, z1, 500, ns, 500, di,
            nullptr, nullptr, nullptr);
        colstats<<<500, 256, 0, stream>>>(z1, ns, 500, bn1g, bn1b, sc1, sh1);

        dim3 g2(cdiv(128, BN), cdiv(ns, BM));
        gemm_wmma_f32<true, false, false><<<g2, 256, 0, stream>>>(
            z1, 500, e2w, 128, e2b, z2, 128, ns, 128, 500,
            sc1, sh1, nullptr);
        colstats<<<128, 256, 0, stream>>>(z2, ns, 128, bn2g, bn2b, sc2, sh2);

        dim3 g3(cdiv(DECD, BN), cdiv(ns, BM));
        gemm_wmma_f32<true, false, false><<<g3, 256, 0, stream>>>(
            z2, 128, dw, DECD, db, dec[m], DECD, ns, DECD, 128,
            sc2, sh2, nullptr);
        colmedian<<<DECD, 256, 0, stream>>>(dec[m], ns, DECD, med + 256 * m);
    }

    // ---- impute + mean over modalities -> nf [N, 256]
    const int* ri0 = (const int*)d_in[11];
    const int* ri1 = (const int*)d_in[23];
    const int* ri2 = (const int*)d_in[35];
    impute_sum<<<cdiv(N_NODES * DECD, 256), 256, 0, stream>>>(
        dec[0], dec[1], dec[2], med, ri0, ri1, ri2, NS[0], NS[1], NS[2], nf);

    // ---- degrees + normalizers
    const int* src = (const int*)d_in[40];
    const int* dst = (const int*)d_in[41];
    zero_f32<<<cdiv(N_NODES, 256), 256, 0, stream>>>(dego, N_NODES);
    zero_f32<<<cdiv(N_NODES, 256), 256, 0, stream>>>(degi, N_NODES);
    degree_count<<<cdiv(E_EDGES, 256), 256, 0, stream>>>(src, E_EDGES, dego);
    degree_count<<<cdiv(E_EDGES, 256), 256, 0, stream>>>(dst, E_EDGES, degi);
    inv_sqrt_clip<<<cdiv(N_NODES, 256), 256, 0, stream>>>(dego, invo, N_NODES);
    inv_sqrt_clip<<<cdiv(N_NODES, 256), 256, 0, stream>>>(degi, invi, N_NODES);

    const float* gw1 = (const float*)d_in[36];
    const float* gb1 = (const float*)d_in[37];
    const float* gw2 = (const float*)d_in[38];
    const float* gb2 = (const float*)d_in[39];

    // ---- gconv1: scatter(nf * invo) -> msg; h = relu((msg*invi) @ gw1 + gb1)
    zero_f32<<<cdiv(N_NODES * DECD, 256), 256, 0, stream>>>(msg, N_NODES * DECD);
    scatter_edges<<<cdiv(E_EDGES, 4), 256, 0, stream>>>(nf, invo, src, dst, msg);
    dim3 gg1(cdiv(DECD, BN), cdiv(N_NODES, BM));
    gemm_wmma_f32<false, true, true><<<gg1, 256, 0, stream>>>(
        msg, DECD, gw1, DECD, gb1, hbuf, DECD, N_NODES, DECD, DECD,
        nullptr, nullptr, invi);

    // ---- gconv2: scatter(h * invo) -> msg; out = (msg*invi) @ gw2 + gb2
    zero_f32<<<cdiv(N_NODES * DECD, 256), 256, 0, stream>>>(msg, N_NODES * DECD);
    scatter_edges<<<cdiv(E_EDGES, 4), 256, 0, stream>>>(hbuf, invo, src, dst, msg);
    dim3 gg2(cdiv(NCLS, BN), cdiv(N_NODES, BM));
    gemm_wmma_f32<false, true, false><<<gg2, 256, 0, stream>>>(
        msg, DECD, gw2, NCLS, gb2, (float*)d_out, NCLS, N_NODES, NCLS, DECD,
        nullptr, nullptr, invi);
}